// GraphTransformer_37993280700519
// MI455X (gfx1250) — compile-verified
//
#include <hip/hip_runtime.h>
#include <hip/hip_bf16.h>

// ---------------- problem constants (match reference) ----------------
#define NN 50000      // nodes
#define EE 600000     // edges
#define DD 128        // features == channels (H=1)
#define GG 64         // graphs
#define NCLS 10       // classes
#define NLAYER 2

typedef float v2f __attribute__((ext_vector_type(2)));
typedef float v8f __attribute__((ext_vector_type(8)));

// =====================================================================
// GEMM: out[n,128] = X[n,128] @ W[128,128] + b[128]
// One kernel computes Q,K,V,S (blockIdx.y selects weight/bias/out).
// Block = 128 threads = 4 waves. Wave w computes column tiles {w, w+4}
// (2 accumulators) for the 16-row strip row0 = 16*blockIdx.x.
// Inner loop: 32 steps of V_WMMA_F32_16X16X4_F32 per accumulator.
// Grid.x = 50000/16 = 3125 exactly -> no divergence, EXEC all 1s.
// =====================================================================
__global__ __launch_bounds__(128)
void gemm_qkvs_kernel(const float* __restrict__ X,
                      const float* __restrict__ Wq, const float* __restrict__ bq, float* __restrict__ Q,
                      const float* __restrict__ Wk, const float* __restrict__ bk, float* __restrict__ K,
                      const float* __restrict__ Wv, const float* __restrict__ bv, float* __restrict__ V,
                      const float* __restrict__ Ws, const float* __restrict__ bs, float* __restrict__ S)
{
    const float* W;  const float* bias;  float* out;
    switch (blockIdx.y) {
        case 0:  W = Wq; bias = bq; out = Q; break;
        case 1:  W = Wk; bias = bk; out = K; break;
        case 2:  W = Wv; bias = bv; out = V; break;
        default: W = Ws; bias = bs; out = S; break;
    }

    const int wave = threadIdx.x >> 5;      // 0..3
    const int lane = threadIdx.x & 31;
    const int lm   = lane & 15;             // 0..15
    const int hi   = lane >> 4;             // 0 or 1 (upper half-wave)
    const int row0 = blockIdx.x * 16;
    const int n0a  = wave * 16;             // first column tile
    const int n0b  = (wave + 4) * 16;       // second column tile

    // C init = bias broadcast along M
    const float ba = bias[n0a + lm];
    const float bb = bias[n0b + lm];
    v8f c0, c1;
#pragma unroll
    for (int r = 0; r < 8; ++r) { c0[r] = ba; c1[r] = bb; }

    const float* xrow = X + (size_t)(row0 + lm) * DD;

#pragma unroll 4
    for (int k = 0; k < DD; k += 4) {
        const int koff = k + 2 * hi;
        // A fragment: row (row0+lm), K = koff, koff+1  (8B-aligned float2)
        const float2 af = *(const float2*)&xrow[koff];
        v2f a; a.x = af.x; a.y = af.y;
        // B fragments: B[koff][n], B[koff+1][n]
        v2f b0, b1;
        b0.x = W[(size_t)koff       * DD + n0a + lm];
        b0.y = W[(size_t)(koff + 1) * DD + n0a + lm];
        b1.x = W[(size_t)koff       * DD + n0b + lm];
        b1.y = W[(size_t)(koff + 1) * DD + n0b + lm];
        c0 = __builtin_amdgcn_wmma_f32_16x16x4_f32(false, a, false, b0, (short)0, c0, false, false);
        c1 = __builtin_amdgcn_wmma_f32_16x16x4_f32(false, a, false, b1, (short)0, c1, false, false);
    }

    // Store C: vgpr r -> row (row0 + r + 8*hi), col n0+lm
#pragma unroll
    for (int r = 0; r < 8; ++r) {
        const int m = row0 + r + 8 * hi;
        out[(size_t)m * DD + n0a + lm] = c0[r];
        out[(size_t)m * DD + n0b + lm] = c1[r];
    }
}

// =====================================================================
// Per-layer node init: running max = -FLT_MAX, denom = 0
// =====================================================================
__global__ __launch_bounds__(256)
void init_nodes_kernel(float* __restrict__ m, float* __restrict__ den)
{
    int i = blockIdx.x * blockDim.x + threadIdx.x;
    if (i < NN) { m[i] = -3.402823466e38f; den[i] = 0.0f; }
}

// =====================================================================
// Edge scores: one wave per edge; lane handles 4 contiguous floats
// score[e] = dot(q[dst], k[src]) / sqrt(128); atomicMax into m[dst]
// =====================================================================
__global__ __launch_bounds__(256)
void edge_score_kernel(const float* __restrict__ q, const float* __restrict__ kb,
                       const int* __restrict__ src, const int* __restrict__ dst,
                       float* __restrict__ score, float* __restrict__ mbuf)
{
    const int gid  = blockIdx.x * blockDim.x + threadIdx.x;
    const int e    = gid >> 5;
    const int lane = gid & 31;
    if (e >= EE) return;
    const int sN = src[e], dN = dst[e];
    const float4 qv = *(const float4*)&q [(size_t)dN * DD + lane * 4];
    const float4 kv = *(const float4*)&kb[(size_t)sN * DD + lane * 4];
    float p = qv.x * kv.x + qv.y * kv.y + qv.z * kv.z + qv.w * kv.w;
#pragma unroll
    for (int off = 16; off > 0; off >>= 1)
        p += __shfl_xor(p, off, 32);
    p *= 0.08838834764831845f;   // 1/sqrt(128)
    if (lane == 0) {
        score[e] = p;
        atomicMax(&mbuf[dN], p);
    }
}

// =====================================================================
// ex[e] = exp(score - m[dst]); den[dst] += ex
// =====================================================================
__global__ __launch_bounds__(256)
void edge_softmax_kernel(float* __restrict__ score, const float* __restrict__ mbuf,
                         float* __restrict__ den, const int* __restrict__ dst)
{
    int e = blockIdx.x * blockDim.x + threadIdx.x;
    if (e >= EE) return;
    const int dN = dst[e];
    const float ev = __expf(score[e] - mbuf[dN]);
    score[e] = ev;
    atomicAdd(&den[dN], ev);
}

// =====================================================================
// s[dst] += (ex/den[dst]) * v[src]   (wave per edge, float atomics)
// s already holds X@Ws + bs, so this accumulates the skip-fused output.
// =====================================================================
__global__ __launch_bounds__(256)
void edge_agg_kernel(const float* __restrict__ score, const float* __restrict__ den,
                     const float* __restrict__ vb,
                     const int* __restrict__ src, const int* __restrict__ dst,
                     float* __restrict__ sbuf)
{
    const int gid  = blockIdx.x * blockDim.x + threadIdx.x;
    const int e    = gid >> 5;
    const int lane = gid & 31;
    if (e >= EE) return;
    const int sN = src[e], dN = dst[e];
    const float alpha = score[e] / den[dN];
    const float4 vv = *(const float4*)&vb[(size_t)sN * DD + lane * 4];
    float* o = &sbuf[(size_t)dN * DD + lane * 4];
    atomicAdd(o + 0, alpha * vv.x);
    atomicAdd(o + 1, alpha * vv.y);
    atomicAdd(o + 2, alpha * vv.z);
    atomicAdd(o + 3, alpha * vv.w);
}

// =====================================================================
// h = relu(s)
// =====================================================================
__global__ __launch_bounds__(256)
void relu_kernel(const float* __restrict__ s, float* __restrict__ h)
{
    int i = blockIdx.x * blockDim.x + threadIdx.x;
    if (i < NN * DD) h[i] = fmaxf(s[i], 0.0f);
}

// =====================================================================
// pool init + accumulate (wave per node)
// =====================================================================
__global__ __launch_bounds__(256)
void init_pool_kernel(float* __restrict__ pooled, float* __restrict__ cnts)
{
    int i = blockIdx.x * blockDim.x + threadIdx.x;
    if (i < GG * DD) pooled[i] = 0.0f;
    if (i < GG)      cnts[i]   = 0.0f;
}

__global__ __launch_bounds__(256)
void pool_kernel(const float* __restrict__ h, const int* __restrict__ batch,
                 float* __restrict__ pooled, float* __restrict__ cnts)
{
    const int gid  = blockIdx.x * blockDim.x + threadIdx.x;
    const int node = gid >> 5;
    const int lane = gid & 31;
    if (node >= NN) return;
    const int g = batch[node];
    const float4 hv = *(const float4*)&h[(size_t)node * DD + lane * 4];
    float* o = &pooled[(size_t)g * DD + lane * 4];
    atomicAdd(o + 0, hv.x);
    atomicAdd(o + 1, hv.y);
    atomicAdd(o + 2, hv.z);
    atomicAdd(o + 3, hv.w);
    if (lane == 0) atomicAdd(&cnts[g], 1.0f);
}

// =====================================================================
// head: logits = (pooled/cnt) @ W_fc + b_fc ; log_softmax
// one thread per graph (64 threads, trivial work)
// =====================================================================
__global__ __launch_bounds__(64)
void head_kernel(const float* __restrict__ pooled, const float* __restrict__ cnts,
                 const float* __restrict__ Wfc, const float* __restrict__ bfc,
                 float* __restrict__ out)
{
    const int g = threadIdx.x;
    if (g >= GG) return;
    const float inv = 1.0f / fmaxf(cnts[g], 1.0f);
    float logit[NCLS];
#pragma unroll
    for (int c = 0; c < NCLS; ++c) logit[c] = bfc[c];
    for (int f = 0; f < DD; ++f) {
        const float pv = pooled[(size_t)g * DD + f] * inv;
#pragma unroll
        for (int c = 0; c < NCLS; ++c) logit[c] += pv * Wfc[f * NCLS + c];
    }
    float mx = logit[0];
#pragma unroll
    for (int c = 1; c < NCLS; ++c) mx = fmaxf(mx, logit[c]);
    float ssum = 0.0f;
#pragma unroll
    for (int c = 0; c < NCLS; ++c) ssum += __expf(logit[c] - mx);
    const float lse = mx + __logf(ssum);
#pragma unroll
    for (int c = 0; c < NCLS; ++c) out[g * NCLS + c] = logit[c] - lse;
}

// =====================================================================
// launch
// =====================================================================
extern "C" void kernel_launch(void* const* d_in, const int* in_sizes, int n_in,
                              void* d_out, int out_size, void* d_ws, size_t ws_size,
                              hipStream_t stream)
{
    const float* x      = (const float*)d_in[0];
    const int*   eidx   = (const int*)  d_in[1];
    const int*   src    = eidx;          // edge_index[0]
    const int*   dst    = eidx + EE;     // edge_index[1]
    const int*   batch  = (const int*)  d_in[2];
    const float* Wq = (const float*)d_in[3];  const float* bq = (const float*)d_in[4];
    const float* Wk = (const float*)d_in[5];  const float* bk = (const float*)d_in[6];
    const float* Wv = (const float*)d_in[7];  const float* bv = (const float*)d_in[8];
    const float* Ws = (const float*)d_in[9];  const float* bs = (const float*)d_in[10];
    const float* Wfc = (const float*)d_in[11];
    const float* bfc = (const float*)d_in[12];
    float* out = (float*)d_out;

    // ---- workspace carve-up (floats) ----
    float* ws  = (float*)d_ws;
    float* h   = ws;                         // N*D
    float* q   = h   + (size_t)NN * DD;      // N*D
    float* kb  = q   + (size_t)NN * DD;      // N*D
    float* vb  = kb  + (size_t)NN * DD;      // N*D
    float* sb  = vb  + (size_t)NN * DD;      // N*D  (skip + aggregate)
    float* ex  = sb  + (size_t)NN * DD;      // E    (score, then exp)
    float* mbf = ex  + (size_t)EE;           // N
    float* den = mbf + (size_t)NN;           // N
    float* pooled = den + (size_t)NN;        // G*D
    float* cnts   = pooled + (size_t)GG * DD; // G

    const dim3 gemmGrid(NN / 16, 4);                 // 3125 x 4, exact
    const int  edgeWaveBlocks = (EE * 32) / 256;     // 75000, exact
    const int  edgeThrBlocks  = (EE + 255) / 256;
    const int  nodeBlocks     = (NN + 255) / 256;
    const int  eltBlocks      = (NN * DD) / 256;     // 25000, exact
    const int  poolWaveBlocks = (NN * 32) / 256;     // 6250, exact

    for (int l = 0; l < NLAYER; ++l) {
        const float* in = (l == 0) ? x : h;
        const size_t wOff = (size_t)l * DD * DD;
        const size_t bOff = (size_t)l * DD;

        init_nodes_kernel<<<nodeBlocks, 256, 0, stream>>>(mbf, den);

        gemm_qkvs_kernel<<<gemmGrid, 128, 0, stream>>>(
            in,
            Wq + wOff, bq + bOff, q,
            Wk + wOff, bk + bOff, kb,
            Wv + wOff, bv + bOff, vb,
            Ws + wOff, bs + bOff, sb);

        edge_score_kernel<<<edgeWaveBlocks, 256, 0, stream>>>(q, kb, src, dst, ex, mbf);
        edge_softmax_kernel<<<edgeThrBlocks, 256, 0, stream>>>(ex, mbf, den, dst);
        edge_agg_kernel<<<edgeWaveBlocks, 256, 0, stream>>>(ex, den, vb, src, dst, sb);
        relu_kernel<<<eltBlocks, 256, 0, stream>>>(sb, h);
    }

    init_pool_kernel<<<(GG * DD + 255) / 256, 256, 0, stream>>>(pooled, cnts);
    pool_kernel<<<poolWaveBlocks, 256, 0, stream>>>(h, batch, pooled, cnts);
    head_kernel<<<1, 64, 0, stream>>>(pooled, cnts, Wfc, bfc, out);
}